// GNNRefinement_48515950576556
// MI455X (gfx1250) — compile-verified
//
#include <hip/hip_runtime.h>
#include <hip/hip_bf16.h>

// ---------------------------------------------------------------------------
// GNN refinement: 2 x (SAGEConv(mean) + LayerNorm + ReLU + residual)
// B=32, C=160, H=W=64  ->  N = 131072 nodes, E=16384 edges (replicated / batch)
//
// fp32 end-to-end; GEMMs run on V_WMMA_F32_16X16X4_F32 (exact fp32 matrix op).
// A-tiles are streamed into LDS with gfx1250 async global->LDS copies.
// ---------------------------------------------------------------------------

typedef float v2f __attribute__((ext_vector_type(2)));
typedef float v8f __attribute__((ext_vector_type(8)));
typedef int   v4i __attribute__((vector_size(4 * sizeof(int))));   // matches builtin proto
typedef __attribute__((address_space(1))) v4i* gv4i_p;             // global int4*
typedef __attribute__((address_space(3))) v4i* lv4i_p;             // LDS int4*

#define GNN_B   32
#define GNN_C   160
#define GNN_K2  320            // combined K: [agg | h] @ [w_l ; w_r]^T
#define GNN_HW  4096           // nodes per batch image
#define GNN_N   (GNN_B * GNN_HW)   // 131072
#define GNN_E   16384
#define GNN_AS  324            // A-tile row stride: 16B-aligned rows (B128 async),
                               // banks (4*col + k) % 64 distinct across the wave
#define GNN_LNS 161            // LDS layernorm-tile row stride

// ---------------------------------------------------------------------------
// gfx1250 async global->LDS b128 copy (ASYNCcnt-tracked), builtin if present,
// else inline asm per CDNA5 ISA.
__device__ __forceinline__ void async_copy_b128(const float* gptr, float* lptr) {
#if __has_builtin(__builtin_amdgcn_global_load_async_to_lds_b128)
    __builtin_amdgcn_global_load_async_to_lds_b128(
        (gv4i_p)(unsigned long long)(uintptr_t)gptr,
        (lv4i_p)(unsigned)(uintptr_t)lptr,
        0, 0);
#else
    const unsigned lofs = (unsigned)(uintptr_t)lptr;   // LDS aperture: addr[31:0]
    asm volatile("global_load_async_to_lds_b128 %0, %1, off"
                 :: "v"(lofs), "v"(gptr) : "memory");
#endif
}

__device__ __forceinline__ void wait_asynccnt0() {
#if __has_builtin(__builtin_amdgcn_s_wait_asynccnt)
    __builtin_amdgcn_s_wait_asynccnt(0);
#else
    asm volatile("s_wait_asynccnt 0x0" ::: "memory");
#endif
}

// ---------------------------------------------------------------------------
// [B,C,H,W] -> [N,C]   (32x32 LDS tile transpose, coalesced loads and stores)
// grid: (4096/32, 160/32, 32), block: (32,8)
__global__ __launch_bounds__(256)
void k_nchw_to_nc(const float* __restrict__ in, float* __restrict__ out) {
    __shared__ float t[32][33];
    const int s0 = blockIdx.x * 32, c0 = blockIdx.y * 32, b = blockIdx.z;
    const int tx = threadIdx.x, ty = threadIdx.y;
#pragma unroll
    for (int j = 0; j < 4; ++j)
        t[ty + 8 * j][tx] = in[((size_t)(b * GNN_C + c0 + ty + 8 * j)) * GNN_HW + s0 + tx];
    __syncthreads();
#pragma unroll
    for (int j = 0; j < 4; ++j)
        out[((size_t)(b * GNN_HW) + s0 + ty + 8 * j) * GNN_C + c0 + tx] = t[tx][ty + 8 * j];
}

// [N,C] -> [B,C,H,W]
__global__ __launch_bounds__(256)
void k_nc_to_nchw(const float* __restrict__ in, float* __restrict__ out) {
    __shared__ float t[32][33];
    const int s0 = blockIdx.x * 32, c0 = blockIdx.y * 32, b = blockIdx.z;
    const int tx = threadIdx.x, ty = threadIdx.y;
#pragma unroll
    for (int j = 0; j < 4; ++j)
        t[ty + 8 * j][tx] = in[((size_t)(b * GNN_HW) + s0 + ty + 8 * j) * GNN_C + c0 + tx];
    __syncthreads();
#pragma unroll
    for (int j = 0; j < 4; ++j)
        out[((size_t)(b * GNN_C + c0 + ty + 8 * j)) * GNN_HW + s0 + tx] = t[tx][ty + 8 * j];
}

// ---------------------------------------------------------------------------
// Degree histogram over dst (same for every batch image -> 4096 bins).
__global__ __launch_bounds__(256)
void k_degree(const int* __restrict__ ei, float* __restrict__ deg) {
    const int i = blockIdx.x * blockDim.x + threadIdx.x;
    if (i < GNN_E) atomicAdd(&deg[ei[GNN_E + i]], 1.0f);
}

// ---------------------------------------------------------------------------
// Pack [w_l ; w_r] transposed -> wt[k * 160 + o]  (k-major: B-frag loads are
// lane-coalesced; 200 KB stays resident in the 192 MB L2 for all 8192 tiles).
__global__ __launch_bounds__(256)
void k_wprep(const float* __restrict__ wl, const float* __restrict__ wr,
             float* __restrict__ wt) {
    const int idx = blockIdx.x * blockDim.x + threadIdx.x;   // [0, 320*160)
    if (idx >= GNN_K2 * GNN_C) return;
    const int k = idx / GNN_C, o = idx % GNN_C;
    wt[idx] = (k < GNN_C) ? wl[o * GNN_C + k] : wr[o * GNN_C + (k - GNN_C)];
}

// ---------------------------------------------------------------------------
// Scatter-sum: agg[dst] += h[src]   (one thread per (edge, channel)).
__global__ __launch_bounds__(256)
void k_aggregate(const float* __restrict__ h, const int* __restrict__ ei,
                 float* __restrict__ agg) {
    const long long idx = (long long)blockIdx.x * blockDim.x + threadIdx.x;
    if (idx >= (long long)GNN_B * GNN_E * GNN_C) return;
    const int c  = (int)(idx % GNN_C);
    const long long e = idx / GNN_C;
    const int b  = (int)(e / GNN_E);
    const int em = (int)(e % GNN_E);
    const int src = ei[em]          + b * GNN_HW;
    const int dst = ei[GNN_E + em]  + b * GNN_HW;
    atomicAdd(&agg[(size_t)dst * GNN_C + c], h[(size_t)src * GNN_C + c]);
}

// ---------------------------------------------------------------------------
// Fused layer. 320 threads = 10 wave32s; wave w owns output columns [16w,16w+16).
// Two independent WMMA chains:  acc1 = agg_tile @ w_l^T,  acc2 = h_tile @ w_r^T;
// the mean scaling 1/deg is applied per-row in the epilogue (commutes with GEMM),
// which makes the A-tile a pure contiguous copy -> async global->LDS streaming.
__global__ __launch_bounds__(320)
void k_layer(const float* __restrict__ hin, const float* __restrict__ agg,
             const float* __restrict__ deg, const float* __restrict__ wt,
             const float* __restrict__ bias, const float* __restrict__ gamma,
             const float* __restrict__ beta, float* __restrict__ hout) {
    __shared__ float Atile[16 * GNN_AS];   // [m][k]  k<160: raw agg, k>=160: h
    __shared__ float Ln[16 * GNN_LNS];     // pre-LN activations
    __shared__ float invdeg_s[16];

    const int tid  = threadIdx.x;
    const int base = blockIdx.x * 16;

    if (tid < 16) {
        const float d = deg[(base + tid) & (GNN_HW - 1)];
        invdeg_s[tid] = 1.0f / fmaxf(d, 1.0f);
    }

    // Async-stream the A-tile: two contiguous 10 KB spans (16 rows x 160 ch),
    // scattered into the bank-padded LDS layout via per-lane LDS addresses.
    // 640 b128 groups per span, 320 threads -> exactly 2 per thread per span.
    {
        const float* gsrc0 = agg + (size_t)base * GNN_C;
        const float* gsrc1 = hin + (size_t)base * GNN_C;
#pragma unroll
        for (int it = 0; it < 2; ++it) {
            const int g  = tid + it * 320;          // group id [0,640)
            const int m  = g / 40;
            const int k4 = (g % 40) * 4;
            async_copy_b128(gsrc0 + g * 4, &Atile[m * GNN_AS + k4]);
            async_copy_b128(gsrc1 + g * 4, &Atile[m * GNN_AS + GNN_C + k4]);
        }
    }
    wait_asynccnt0();
    __syncthreads();

    const int wv   = tid >> 5;
    const int lane = tid & 31;
    const int half = lane >> 4;        // 0: K+0/1, 1: K+2/3 (ISA 7.12.2 f32 layout)
    const int col  = lane & 15;
    const int co   = wv * 16;

    const float* bp = wt + (co + col); // column base; element k at bp[k*160]
    __builtin_prefetch(bp, 0, 1);      // global_prefetch_b8: warm weight column

    v8f acc1 = {};   // agg @ w_l^T
    v8f acc2 = {};   // h   @ w_r^T
#pragma unroll 4
    for (int k0 = 0; k0 < GNN_C; k0 += 4) {
        const int kk = k0 + 2 * half;
        // A frags: row = lane&15; agg part and h part 640 B apart
        // (merges into ds_load_2addr_b64).
        const v2f a1 = *(const v2f*)&Atile[col * GNN_AS + kk];
        const v2f a2 = *(const v2f*)&Atile[col * GNN_AS + GNN_C + kk];
        v2f b1, b2;
        b1.x = bp[(size_t)kk * GNN_C];
        b1.y = bp[(size_t)(kk + 1) * GNN_C];
        b2.x = bp[(size_t)(GNN_C + kk) * GNN_C];
        b2.y = bp[(size_t)(GNN_C + kk + 1) * GNN_C];
        acc1 = __builtin_amdgcn_wmma_f32_16x16x4_f32(
                   false, a1, false, b1, (short)0, acc1, false, false);
        acc2 = __builtin_amdgcn_wmma_f32_16x16x4_f32(
                   false, a2, false, b2, (short)0, acc2, false, false);
    }

    // Epilogue: mean-scale the agg chain per row, +bias, spill to LDS for LN.
    const float bcol = bias[co + col];
#pragma unroll
    for (int r = 0; r < 8; ++r) {
        const int m = r + 8 * half;    // C/D layout: vgpr r -> rows r / r+8
        Ln[m * GNN_LNS + co + col] = acc1[r] * invdeg_s[m] + acc2[r] + bcol;
    }
    __syncthreads();

    // LayerNorm + ReLU + residual: 8 waves x 2 nodes, wave32 shfl reductions.
    if (wv < 8) {
#pragma unroll
        for (int sub = 0; sub < 2; ++sub) {
            const int m = wv * 2 + sub;
            float vals[5], s = 0.f, ss = 0.f;
#pragma unroll
            for (int i = 0; i < 5; ++i) {
                const float v = Ln[m * GNN_LNS + lane + 32 * i];
                vals[i] = v; s += v; ss += v * v;
            }
#pragma unroll
            for (int off = 16; off > 0; off >>= 1) {
                s  += __shfl_xor(s,  off, 32);
                ss += __shfl_xor(ss, off, 32);
            }
            const float mu   = s * (1.0f / GNN_C);
            const float var  = ss * (1.0f / GNN_C) - mu * mu;
            const float rstd = rsqrtf(var + 1e-5f);
            const size_t n = (size_t)(base + m);
#pragma unroll
            for (int i = 0; i < 5; ++i) {
                const int c = lane + 32 * i;
                float v = (vals[i] - mu) * rstd * gamma[c] + beta[c];
                v = fmaxf(v, 0.0f);                          // ReLU
                hout[n * GNN_C + c] = Atile[m * GNN_AS + GNN_C + c] + v; // residual
            }
        }
    }
}

// ---------------------------------------------------------------------------
extern "C" void kernel_launch(void* const* d_in, const int* in_sizes, int n_in,
                              void* d_out, int out_size, void* d_ws, size_t ws_size,
                              hipStream_t stream) {
    const float* spat = (const float*)d_in[0];
    const int*   ei   = (const int*)d_in[1];
    const float* w1l  = (const float*)d_in[2];
    const float* w1r  = (const float*)d_in[3];
    const float* b1   = (const float*)d_in[4];
    const float* g1   = (const float*)d_in[5];
    const float* be1  = (const float*)d_in[6];
    const float* w2l  = (const float*)d_in[7];
    const float* w2r  = (const float*)d_in[8];
    const float* b2   = (const float*)d_in[9];
    const float* g2   = (const float*)d_in[10];
    const float* be2  = (const float*)d_in[11];
    float* outp = (float*)d_out;

    // Workspace carve-up (floats): h0 | h1 | agg | deg | wt1 | wt2
    const size_t NC = (size_t)GNN_N * GNN_C;          // 20,971,520 floats
    float* h0  = (float*)d_ws;
    float* h1  = h0 + NC;
    float* agg = h1 + NC;
    float* deg = agg + NC;
    float* wt1 = deg + GNN_HW;
    float* wt2 = wt1 + (size_t)GNN_K2 * GNN_C;

    const dim3 tb(32, 8);
    const dim3 tgrid(GNN_HW / 32, GNN_C / 32, GNN_B);

    // Prep: layout transform, degree histogram, packed transposed weights.
    (void)hipMemsetAsync(deg, 0, GNN_HW * sizeof(float), stream);
    k_nchw_to_nc<<<tgrid, tb, 0, stream>>>(spat, h0);
    k_degree<<<(GNN_E + 255) / 256, 256, 0, stream>>>(ei, deg);
    {
        const int nw = GNN_K2 * GNN_C;
        k_wprep<<<(nw + 255) / 256, 256, 0, stream>>>(w1l, w1r, wt1);
        k_wprep<<<(nw + 255) / 256, 256, 0, stream>>>(w2l, w2r, wt2);
    }

    const long long aggWork = (long long)GNN_B * GNN_E * GNN_C;   // 83,886,080
    const int aggBlocks = (int)((aggWork + 255) / 256);
    const int layerBlocks = GNN_N / 16;                           // 8192

    // Layer 1: h0 -> h1
    (void)hipMemsetAsync(agg, 0, NC * sizeof(float), stream);
    k_aggregate<<<aggBlocks, 256, 0, stream>>>(h0, ei, agg);
    k_layer<<<layerBlocks, 320, 0, stream>>>(h0, agg, deg, wt1, b1, g1, be1, h1);

    // Layer 2: h1 -> h0 (reuse), then back to NCHW.
    (void)hipMemsetAsync(agg, 0, NC * sizeof(float), stream);
    k_aggregate<<<aggBlocks, 256, 0, stream>>>(h1, ei, agg);
    k_layer<<<layerBlocks, 320, 0, stream>>>(h1, agg, deg, wt2, b2, g2, be2, h0);

    k_nc_to_nchw<<<tgrid, tb, 0, stream>>>(h0, outp);
}